// SpectralConnectivityLoss_37855841747242
// MI455X (gfx1250) — compile-verified
//
#include <hip/hip_runtime.h>
#include <math.h>
#include <stdint.h>

// ---------------------------------------------------------------------------
// SpectralConnectivityLoss for MI455X (gfx1250, wave32).
//   per batch b:  A = sigmoid(logits[b])           (64x64)
//                 M = sym(diag(rowsum(A)) - A)     (eigvalsh symmetrizes)
//                 lambda_2 = 2nd smallest eig(M)
//   loss = mean(relu(0.1 - lambda_2))
//
// One workgroup per graph; operator lives in LDS. Gershgorin flip
// S = (cI - M)/c, block subspace iteration (k=16) with X <- S*X on
// V_WMMA_F32_16X16X4_F32, Cholesky-QR every 8 iters (Gram matrix via WMMA),
// Rayleigh-Ritz H = X^T M X via WMMA + 16x16 Jacobi for the bottom Ritz
// values.  Input streamed HBM->LDS with GLOBAL_LOAD_ASYNC_TO_LDS_B128
// (ASYNCcnt).
// ---------------------------------------------------------------------------

#define NBATCH   4096
#define NN       64
#define LD       65        // padded row stride for 64x64 LDS matrices
#define KSUB     16        // subspace block width (one WMMA tile)
#define LDX      17        // padded row stride for 64x16 LDS matrices
#define NTHREADS 128       // 4 waves of 32
#define NITER    32
#define EPS_CONN 0.1f

typedef __attribute__((ext_vector_type(2))) float v2f;
typedef __attribute__((ext_vector_type(8))) float v8f;

#if defined(__gfx1250__) && __has_builtin(__builtin_amdgcn_global_load_async_to_lds_b128)
#define USE_ASYNC_LDS 1
// exact pointee type from the builtin's prototype: int __vector_size__(16)
typedef int v4i __attribute__((vector_size(16)));
typedef __attribute__((address_space(1))) v4i as1_v4i;   // global (__device__)
typedef __attribute__((address_space(3))) v4i as3_v4i;   // LDS (__shared__)
#else
#define USE_ASYNC_LDS 0
#endif

#define HAVE_WMMA_F32X4 __has_builtin(__builtin_amdgcn_wmma_f32_16x16x4_f32)

__device__ __forceinline__ void wait_async_lds()
{
#if USE_ASYNC_LDS
#if __has_builtin(__builtin_amdgcn_s_wait_asynccnt)
    __builtin_amdgcn_s_wait_asynccnt(0);
#else
    asm volatile("s_wait_asynccnt 0x0" ::: "memory");
#endif
#endif
}

// --------------------------------------------------------------------------
// Y(64x16) = Amat(64x64, stride LD) * Bmat(64x16, stride LDX).
// Each wave computes one 16x16 tile of Y (tile row = wave id); K=64 in 16
// chunks of 4 with V_WMMA_F32_16X16X4_F32.
//
// Assumed wave32 f32 fragment layouts (ISA 7.12.2):
//   A 16x4 : lane L -> row M = L&15 ; VGPR v -> K = v + 2*(L>>4)
//   B 4x16 : lane L -> col N = L&15 ; VGPR v -> K = v + 2*(L>>4)
//   C/D    : lane L -> col N = L&15 ; VGPR r -> row M = r + 8*(L>>4)
// --------------------------------------------------------------------------
__device__ __forceinline__ void gemm64x16(const float* __restrict__ Amat,
                                          const float* __restrict__ Bmat,
                                          float* __restrict__ Cmat,
                                          int wv, int lane)
{
#if HAVE_WMMA_F32X4
    const int m16   = lane & 15;
    const int hi    = lane >> 4;      // 0 or 1
    const int khalf = hi << 1;        // 0 or 2
    const int row   = wv * 16 + m16;

    v8f acc = {};
    #pragma unroll
    for (int kk = 0; kk < NN / 4; ++kk) {
        const int kb = kk * 4 + khalf;
        v2f a; a.x = Amat[row * LD + kb];        a.y = Amat[row * LD + kb + 1];
        v2f b; b.x = Bmat[(kb + 0) * LDX + m16]; b.y = Bmat[(kb + 1) * LDX + m16];
        acc = __builtin_amdgcn_wmma_f32_16x16x4_f32(
                  false, a, false, b, (short)0, acc, false, false);
    }
    #pragma unroll
    for (int r = 0; r < 8; ++r)
        Cmat[(wv * 16 + r + (hi << 3)) * LDX + m16] = acc[r];
#else
    for (int e = lane; e < 16 * KSUB; e += 32) {
        const int m = wv * 16 + (e >> 4), n = e & 15;
        float s = 0.f;
        for (int k = 0; k < NN; ++k) s += Amat[m * LD + k] * Bmat[k * LDX + n];
        Cmat[m * LDX + n] = s;
    }
#endif
}

// --------------------------------------------------------------------------
// C(16x16) = X^T(16x64) * W(64x16), both stride LDX; C stored dense 16x16.
// Since A[m][k] = X[k][m], the A-fragment has the same per-lane access
// pattern as a B-fragment (column m of X).  Run by ONE wave (wave-uniform
// branch -> EXEC all-1s inside the wave, as WMMA requires).
// --------------------------------------------------------------------------
__device__ __forceinline__ void gemm16x16_AtB(const float* __restrict__ X,
                                              const float* __restrict__ W,
                                              float* __restrict__ C, int lane)
{
#if HAVE_WMMA_F32X4
    const int m16   = lane & 15;
    const int hi    = lane >> 4;
    const int khalf = hi << 1;

    v8f acc = {};
    #pragma unroll
    for (int kk = 0; kk < NN / 4; ++kk) {
        const int kb = kk * 4 + khalf;
        v2f a; a.x = X[(kb + 0) * LDX + m16]; a.y = X[(kb + 1) * LDX + m16];
        v2f b; b.x = W[(kb + 0) * LDX + m16]; b.y = W[(kb + 1) * LDX + m16];
        acc = __builtin_amdgcn_wmma_f32_16x16x4_f32(
                  false, a, false, b, (short)0, acc, false, false);
    }
    #pragma unroll
    for (int r = 0; r < 8; ++r)
        C[(r + (hi << 3)) * KSUB + m16] = acc[r];
#else
    for (int e = lane; e < KSUB * KSUB; e += 32) {
        const int i = e >> 4, j = e & 15;
        float s = 0.f;
        for (int k = 0; k < NN; ++k) s += X[k * LDX + i] * W[k * LDX + j];
        C[e] = s;
    }
#endif
}

// --------------------------------------------------------------------------
// Cholesky-QR re-orthonormalization of X (64x16): G = X^T X = L L^T,
// X <- X * L^{-T}.  G via WMMA on wave 0.
// --------------------------------------------------------------------------
__device__ void cholesky_qr(float* __restrict__ X, float* __restrict__ G,
                            float* __restrict__ Linv, float* __restrict__ Ytmp,
                            int t)
{
    const int lane = t & 31, wv = t >> 5;

    if (wv == 0) gemm16x16_AtB(X, X, G, lane);       // G = X^T X (WMMA)
    __syncthreads();

    if (t == 0) {
        for (int j = 0; j < KSUB; ++j) {             // lower Cholesky
            float d = G[j * KSUB + j] + 1e-12f;
            for (int k = 0; k < j; ++k) d -= G[j * KSUB + k] * G[j * KSUB + k];
            d = sqrtf(fmaxf(d, 1e-20f));
            G[j * KSUB + j] = d;
            const float inv = 1.0f / d;
            for (int i = j + 1; i < KSUB; ++i) {
                float s = G[i * KSUB + j];
                for (int k = 0; k < j; ++k) s -= G[i * KSUB + k] * G[j * KSUB + k];
                G[i * KSUB + j] = s * inv;
            }
        }
        for (int i = 0; i < KSUB; ++i) {             // invert L (lower)
            Linv[i * KSUB + i] = 1.0f / G[i * KSUB + i];
            for (int j = i - 1; j >= 0; --j) {
                float s = 0.f;
                for (int k = j + 1; k <= i; ++k)
                    s += G[k * KSUB + j] * Linv[i * KSUB + k];
                Linv[i * KSUB + j] = -s / G[j * KSUB + j];
            }
            for (int j = i + 1; j < KSUB; ++j) Linv[i * KSUB + j] = 0.f;
        }
    }
    __syncthreads();

    for (int e = t; e < NN * KSUB; e += NTHREADS) {  // X <- X * L^{-T}
        const int r = e >> 4, j = e & 15;
        float s = 0.f;
        for (int k = 0; k <= j; ++k) s += X[r * LDX + k] * Linv[j * KSUB + k];
        Ytmp[r * LDX + j] = s;
    }
    __syncthreads();
    for (int e = t; e < NN * KSUB; e += NTHREADS) {
        const int r = e >> 4, j = e & 15;
        X[r * LDX + j] = Ytmp[r * LDX + j];
    }
    __syncthreads();
}

// --------------------------------------------------------------------------
// Main kernel: one block per graph; writes per-batch penalty to pen[].
// --------------------------------------------------------------------------
__global__ __launch_bounds__(NTHREADS)
void spectral_lambda2_kernel(const float* __restrict__ logits,
                             float* __restrict__ pen)
{
    __shared__ __align__(16) float Sbuf[NN * LD];   // A, then flipped op S
    __shared__ __align__(16) float Mbuf[NN * LD];   // staging, then sym Laplacian
    __shared__ float Xbuf[NN * LDX];                // subspace block
    __shared__ float Ybuf[NN * LDX];                // product / scratch
    __shared__ float deg[NN];
    __shared__ float rowrad[NN];
    __shared__ float Gm[KSUB * KSUB];
    __shared__ float Linv[KSUB * KSUB];
    __shared__ float Hm[KSUB * KSUB];
    __shared__ float cbound;

    const int t    = threadIdx.x;
    const int lane = t & 31;
    const int wv   = t >> 5;

    const float* src = logits + (size_t)blockIdx.x * (NN * NN);

    // ---- stream raw logits HBM -> LDS (Mbuf reused as staging) ------------
#if USE_ASYNC_LDS
    {
        const uintptr_t gbase = (uintptr_t)src;
        const uint32_t  lbase = (uint32_t)(uintptr_t)(void*)Mbuf;
        #pragma unroll
        for (int i = 0; i < 8; ++i) {
            const int v4 = i * NTHREADS + t;             // 16B granule 0..1023
            __builtin_amdgcn_global_load_async_to_lds_b128(
                (as1_v4i*)(gbase + (uintptr_t)v4 * 16),
                (as3_v4i*)(uintptr_t)(lbase + (uint32_t)v4 * 16),
                0, 0);
        }
        wait_async_lds();                                // ASYNCcnt -> 0
    }
    __syncthreads();
    for (int e = t; e < NN * NN; e += NTHREADS) {        // sigmoid LDS->LDS
        const float x = Mbuf[e];
        Sbuf[(e >> 6) * LD + (e & 63)] = 1.f / (1.f + __expf(-x));
    }
#else
    __builtin_prefetch(src, 0, 1);                       // global_prefetch_b8
    #pragma unroll
    for (int i = 0; i < 8; ++i) {
        const int v4 = i * NTHREADS + t;
        const float4 x = ((const float4*)src)[v4];
        const int base = v4 * 4;
        float* dst = &Sbuf[(base >> 6) * LD + (base & 63)];
        dst[0] = 1.f / (1.f + __expf(-x.x));
        dst[1] = 1.f / (1.f + __expf(-x.y));
        dst[2] = 1.f / (1.f + __expf(-x.z));
        dst[3] = 1.f / (1.f + __expf(-x.w));
    }
#endif
    __syncthreads();

    // ---- degrees ----------------------------------------------------------
    if (t < NN) {
        float s = 0.f;
        for (int j = 0; j < NN; ++j) s += Sbuf[t * LD + j];
        deg[t] = s;
    }
    __syncthreads();

    // ---- M = diag(deg) - (A + A^T)/2  (overwrites staging) ----------------
    for (int e = t; e < NN * NN; e += NTHREADS) {
        const int i = e >> 6, j = e & 63;
        float m = -0.5f * (Sbuf[i * LD + j] + Sbuf[j * LD + i]);
        if (i == j) m += deg[i];
        Mbuf[i * LD + j] = m;
    }
    __syncthreads();

    // ---- Gershgorin bound c, then S = (cI - M)/c --------------------------
    if (t < NN) {
        float s = Mbuf[t * LD + t];
        for (int j = 0; j < NN; ++j)
            if (j != t) s += fabsf(Mbuf[t * LD + j]);
        rowrad[t] = s;
    }
    __syncthreads();
    if (t == 0) {
        float c = 1e-6f;
        for (int i = 0; i < NN; ++i) c = fmaxf(c, rowrad[i]);
        cbound = c;
    }
    __syncthreads();
    const float cb = cbound, cinv = 1.f / cb;
    for (int e = t; e < NN * NN; e += NTHREADS) {
        const int i = e >> 6, j = e & 63;
        float s = -Mbuf[i * LD + j];
        if (i == j) s += cb;
        Sbuf[i * LD + j] = s * cinv;                 // A no longer needed
    }

    // ---- deterministic subspace start -------------------------------------
    for (int e = t; e < NN * KSUB; e += NTHREADS) {
        unsigned h = (unsigned)e * 2654435761u + 12345u;
        h ^= h >> 13; h *= 2246822519u; h ^= h >> 16;
        Xbuf[(e >> 4) * LDX + (e & 15)] =
            (float)(h & 0xFFFF) * (1.f / 65536.f) - 0.5f;
    }
    __syncthreads();

    // ---- block subspace iteration: X <- S*X, ortho every 8 ----------------
    for (int it = 0; it < NITER; ++it) {
        gemm64x16(Sbuf, Xbuf, Ybuf, wv, lane);       // WMMA f32 16x16x4
        __syncthreads();
        for (int e = t; e < NN * KSUB; e += NTHREADS) {
            const int r = e >> 4, j = e & 15;
            Xbuf[r * LDX + j] = Ybuf[r * LDX + j];
        }
        __syncthreads();
        if ((it & 7) == 7)
            cholesky_qr(Xbuf, Gm, Linv, Ybuf, t);
    }

    // ---- Rayleigh-Ritz: W = M X (WMMA), H = X^T W (WMMA, wave 0) ----------
    gemm64x16(Mbuf, Xbuf, Ybuf, wv, lane);
    __syncthreads();
    if (wv == 0) gemm16x16_AtB(Xbuf, Ybuf, Hm, lane);
    __syncthreads();

    // ---- 16x16 Jacobi on H (serial, one lane), take 2nd smallest ----------
    if (t == 0) {
        for (int sweep = 0; sweep < 10; ++sweep) {
            for (int p = 0; p < KSUB - 1; ++p) {
                for (int q = p + 1; q < KSUB; ++q) {
                    const float apq = Hm[p * KSUB + q];
                    if (fabsf(apq) < 1e-12f) continue;
                    const float app = Hm[p * KSUB + p];
                    const float aqq = Hm[q * KSUB + q];
                    const float th  = 0.5f * (aqq - app) / apq;
                    const float tt  = (th >= 0.f ? 1.f : -1.f) /
                                      (fabsf(th) + sqrtf(th * th + 1.f));
                    const float cc  = 1.f / sqrtf(tt * tt + 1.f);
                    const float ss  = tt * cc;
                    for (int k = 0; k < KSUB; ++k) {
                        const float hkp = Hm[k * KSUB + p];
                        const float hkq = Hm[k * KSUB + q];
                        Hm[k * KSUB + p] = cc * hkp - ss * hkq;
                        Hm[k * KSUB + q] = ss * hkp + cc * hkq;
                    }
                    for (int k = 0; k < KSUB; ++k) {
                        const float hpk = Hm[p * KSUB + k];
                        const float hqk = Hm[q * KSUB + k];
                        Hm[p * KSUB + k] = cc * hpk - ss * hqk;
                        Hm[q * KSUB + k] = ss * hpk + cc * hqk;
                    }
                }
            }
        }
        float m1 = 3.4e38f, m2 = 3.4e38f;
        for (int i = 0; i < KSUB; ++i) {
            const float e = Hm[i * KSUB + i];
            if (e < m1)      { m2 = m1; m1 = e; }
            else if (e < m2) { m2 = e; }
        }
        pen[blockIdx.x] = fmaxf(EPS_CONN - m2, 0.f);
    }
}

// --------------------------------------------------------------------------
// Deterministic fixed-order mean reduction of the 4096 penalties.
// --------------------------------------------------------------------------
__global__ void spectral_reduce_kernel(const float* __restrict__ pen,
                                       float* __restrict__ out)
{
    __shared__ float s[256];
    const int t = threadIdx.x;
    float acc = 0.f;
    for (int i = t; i < NBATCH; i += 256) acc += pen[i];
    s[t] = acc;
    __syncthreads();
    for (int w = 128; w > 0; w >>= 1) {
        if (t < w) s[t] += s[t + w];
        __syncthreads();
    }
    if (t == 0) out[0] = s[0] * (1.0f / (float)NBATCH);
}

extern "C" void kernel_launch(void* const* d_in, const int* in_sizes, int n_in,
                              void* d_out, int out_size, void* d_ws, size_t ws_size,
                              hipStream_t stream)
{
    const float* logits = (const float*)d_in[0];   // (4096,64,64) f32
    // d_in[1] = node_types (int64) -- unused by the reference loss
    float* out = (float*)d_out;                    // 1 x f32
    float* pen = (float*)d_ws;                     // 4096 x f32 scratch

    spectral_lambda2_kernel<<<dim3(NBATCH), dim3(NTHREADS), 0, stream>>>(logits, pen);
    spectral_reduce_kernel<<<dim3(1), dim3(256), 0, stream>>>(pen, out);
}